// MPNNWithEdgeFeatures_90039694393774
// MI455X (gfx1250) — compile-verified
//
#include <hip/hip_runtime.h>
#include <hip/hip_bf16.h>

#define HID 64

typedef __attribute__((ext_vector_type(2))) float v2f;
typedef __attribute__((ext_vector_type(8))) float v8f;

__device__ __forceinline__ v8f wmma_f32_k4(v2f a, v2f b, v8f c) {
    // D = A(16x4) * B(4x16) + C(16x16), all f32, wave32 WMMA
    return __builtin_amdgcn_wmma_f32_16x16x4_f32(
        /*neg_a=*/false, a, /*neg_b=*/false, b,
        /*c_mod=*/(short)0, c, /*reuse_a=*/false, /*reuse_b=*/false);
}

// ---------------------------------------------------------------------------
// Degree / normalization
// ---------------------------------------------------------------------------
__global__ void deg_init_kernel(float* __restrict__ deg, int n) {
    int i = blockIdx.x * blockDim.x + threadIdx.x;
    if (i < n) deg[i] = 1.0f;  // self loop contributes 1
}

__global__ void deg_accum_kernel(const int* __restrict__ ei, float* __restrict__ deg, int nE) {
    int e = blockIdx.x * blockDim.x + threadIdx.x;
    if (e < nE) unsafeAtomicAdd(&deg[ei[nE + e]], 1.0f);  // dst side
}

__global__ void dinv_kernel(const float* __restrict__ deg, float* __restrict__ dinv, int n) {
    int i = blockIdx.x * blockDim.x + threadIdx.x;
    if (i < n) {
        float d = deg[i];
        dinv[i] = d > 0.0f ? rsqrtf(d) : 0.0f;
    }
}

// ---------------------------------------------------------------------------
// Source/target index discovery (argmax of boolean == first match == min idx)
// ---------------------------------------------------------------------------
__global__ void idx_init_kernel(int* __restrict__ idx) {
    if (threadIdx.x == 0) { idx[0] = 0x7fffffff; idx[1] = 0x7fffffff; }
}

__global__ void find_idx_kernel(const float* __restrict__ x, int* __restrict__ idx, int n) {
    int i = blockIdx.x * blockDim.x + threadIdx.x;
    if (i >= n) return;
    if (x[i * 4 + 2] == 1.0f) atomicMin(&idx[0], i);
    if (x[i * 4 + 3] == 1.0f) atomicMin(&idx[1], i);
}

// ---------------------------------------------------------------------------
// Node encoder MLP: h = relu(relu(x@W1 + b1)@W2 + b2)    (x: [n,4])
// Block = 128 threads = 4 waves; each wave owns a 16-row tile.
// ---------------------------------------------------------------------------
__global__ void node_mlp_kernel(const float* __restrict__ x,
                                const float* __restrict__ W1, const float* __restrict__ b1,
                                const float* __restrict__ W2, const float* __restrict__ b2,
                                float* __restrict__ h, int n) {
    __shared__ float sW1[4 * HID];
    __shared__ float sb1[HID];
    __shared__ float sW2[HID * HID];
    __shared__ float sb2[HID];
    __shared__ float sTile[4][16 * HID];  // per-wave stage-1 result

    const int tid  = threadIdx.x;
    const int wave = tid >> 5;
    const int lane = tid & 31;
    const int half = lane >> 4;   // 0 or 1
    const int m    = lane & 15;   // row within 16 (A / M index)
    const int nn   = lane & 15;   // col within 16 (B / N index)

    for (int i = tid; i < 4 * HID; i += 128) sW1[i] = W1[i];
    for (int i = tid; i < HID; i += 128) { sb1[i] = b1[i]; sb2[i] = b2[i]; }
    for (int i = tid; i < HID * HID; i += 128) sW2[i] = W2[i];
    __syncthreads();

    const int row0 = (blockIdx.x * 4 + wave) * 16;
    float* tile = sTile[wave];

    // --- stage 1: x(16x4) @ W1(4x64), single K=4 chunk, 4 column tiles ---
    v2f a;
    {
        int r = row0 + m;
        bool ok = r < n;
        a[0] = ok ? x[r * 4 + 2 * half + 0] : 0.0f;  // A layout: K = 2*half + vgpr
        a[1] = ok ? x[r * 4 + 2 * half + 1] : 0.0f;
    }
#pragma unroll
    for (int nb = 0; nb < 4; ++nb) {
        v2f b;
        b[0] = sW1[(2 * half + 0) * HID + nb * 16 + nn];
        b[1] = sW1[(2 * half + 1) * HID + nb * 16 + nn];
        v8f c = {0.f, 0.f, 0.f, 0.f, 0.f, 0.f, 0.f, 0.f};
        c = wmma_f32_k4(a, b, c);
#pragma unroll
        for (int r = 0; r < 8; ++r) {
            int row = r + 8 * half;          // C/D layout
            int col = nb * 16 + nn;
            float v = c[r] + sb1[col];
            tile[row * HID + col] = v > 0.0f ? v : 0.0f;
        }
    }
    // per-wave LDS: DS ops are in-order within a wave, no cross-wave sync needed

    // --- stage 2: tile(16x64) @ W2(64x64), 16 K-chunks, 4 column tiles ---
    v8f acc[4];
#pragma unroll
    for (int nb = 0; nb < 4; ++nb)
#pragma unroll
        for (int r = 0; r < 8; ++r) acc[nb][r] = 0.0f;

    for (int kk = 0; kk < 16; ++kk) {
        v2f a2;
        a2[0] = tile[m * HID + 4 * kk + 2 * half + 0];
        a2[1] = tile[m * HID + 4 * kk + 2 * half + 1];
#pragma unroll
        for (int nb = 0; nb < 4; ++nb) {
            v2f b2v;
            b2v[0] = sW2[(4 * kk + 2 * half + 0) * HID + nb * 16 + nn];
            b2v[1] = sW2[(4 * kk + 2 * half + 1) * HID + nb * 16 + nn];
            acc[nb] = wmma_f32_k4(a2, b2v, acc[nb]);
        }
    }
#pragma unroll
    for (int nb = 0; nb < 4; ++nb)
#pragma unroll
        for (int r = 0; r < 8; ++r) {
            int row = row0 + r + 8 * half;
            if (row < n) {
                int col = nb * 16 + nn;
                float v = acc[nb][r] + sb2[col];
                h[(size_t)row * HID + col] = v > 0.0f ? v : 0.0f;
            }
        }
}

// ---------------------------------------------------------------------------
// GCN linear: hw = h @ W     (bias applied post-aggregation)
// ---------------------------------------------------------------------------
__global__ void gcn_linear_kernel(const float* __restrict__ h, const float* __restrict__ W,
                                  float* __restrict__ hw, int n) {
    __shared__ float sW[HID * HID];
    __shared__ float sT[4][16 * HID];

    const int tid  = threadIdx.x;
    const int wave = tid >> 5;
    const int lane = tid & 31;
    const int half = lane >> 4;
    const int m    = lane & 15;
    const int nn   = lane & 15;

    for (int i = tid; i < HID * HID; i += 128) sW[i] = W[i];

    const int row0 = (blockIdx.x * 4 + wave) * 16;
    float* tile = sT[wave];
    for (int i = lane; i < 16 * HID; i += 32) {
        int rr = row0 + i / HID;
        tile[i] = (rr < n) ? h[(size_t)rr * HID + (i % HID)] : 0.0f;
    }
    __syncthreads();  // sW visibility across waves

    v8f acc[4];
#pragma unroll
    for (int nb = 0; nb < 4; ++nb)
#pragma unroll
        for (int r = 0; r < 8; ++r) acc[nb][r] = 0.0f;

    for (int kk = 0; kk < 16; ++kk) {
        v2f a2;
        a2[0] = tile[m * HID + 4 * kk + 2 * half + 0];
        a2[1] = tile[m * HID + 4 * kk + 2 * half + 1];
#pragma unroll
        for (int nb = 0; nb < 4; ++nb) {
            v2f b2v;
            b2v[0] = sW[(4 * kk + 2 * half + 0) * HID + nb * 16 + nn];
            b2v[1] = sW[(4 * kk + 2 * half + 1) * HID + nb * 16 + nn];
            acc[nb] = wmma_f32_k4(a2, b2v, acc[nb]);
        }
    }
#pragma unroll
    for (int nb = 0; nb < 4; ++nb)
#pragma unroll
        for (int r = 0; r < 8; ++r) {
            int row = row0 + r + 8 * half;
            if (row < n) hw[(size_t)row * HID + nb * 16 + nn] = acc[nb][r];
        }
}

// ---------------------------------------------------------------------------
// agg = hw * dinv^2  (self-loop term; also zero-initializes agg)
// ---------------------------------------------------------------------------
__global__ void selfloop_kernel(const float* __restrict__ hw, const float* __restrict__ dinv,
                                float* __restrict__ agg, int n) {
    long long t = (long long)blockIdx.x * blockDim.x + threadIdx.x;
    if (t >= (long long)n * HID) return;
    int i = (int)(t >> 6);
    float di = dinv[i];
    agg[t] = hw[t] * di * di;
}

// ---------------------------------------------------------------------------
// Edge scatter: agg[dst] += hw[src] * dinv[src]*dinv[dst]
// One thread per (edge, column) -> coalesced 64-wide rows, f32 HW atomics in L2.
// ---------------------------------------------------------------------------
__global__ void scatter_kernel(const float* __restrict__ hw, const int* __restrict__ ei,
                               const float* __restrict__ dinv, float* __restrict__ agg,
                               int nE) {
    long long t = (long long)blockIdx.x * blockDim.x + threadIdx.x;
    if (t >= (long long)nE * HID) return;
    int e   = (int)(t >> 6);
    int col = (int)(t & 63);
    int s = ei[e];
    int d = ei[nE + e];
    float w = dinv[s] * dinv[d];
    unsafeAtomicAdd(&agg[(size_t)d * HID + col], hw[(size_t)s * HID + col] * w);
}

// ---------------------------------------------------------------------------
// h += relu(agg + b)
// ---------------------------------------------------------------------------
__global__ void update_kernel(float* __restrict__ h, const float* __restrict__ agg,
                              const float* __restrict__ b, int n) {
    long long t = (long long)blockIdx.x * blockDim.x + threadIdx.x;
    if (t >= (long long)n * HID) return;
    int col = (int)(t & 63);
    float v = agg[t] + b[col];
    h[t] += v > 0.0f ? v : 0.0f;
}

// ---------------------------------------------------------------------------
// Readout MLP: [h[s], h[t]] (128) -> 64 -> 32 -> 1
// ---------------------------------------------------------------------------
__global__ void readout_kernel(const float* __restrict__ h, const int* __restrict__ idx,
                               const float* __restrict__ W1, const float* __restrict__ b1,
                               const float* __restrict__ W2, const float* __restrict__ b2,
                               const float* __restrict__ W3, const float* __restrict__ b3,
                               float* __restrict__ out) {
    __shared__ float comb[2 * HID];
    __shared__ float o1[HID];
    __shared__ float o2[HID / 2];
    int t = threadIdx.x;
    int s = idx[0] == 0x7fffffff ? 0 : idx[0];
    int d = idx[1] == 0x7fffffff ? 0 : idx[1];
    if (t < HID) comb[t] = h[(size_t)s * HID + t];
    else         comb[t] = h[(size_t)d * HID + (t - HID)];
    __syncthreads();
    if (t < HID) {
        float acc = b1[t];
        for (int k = 0; k < 2 * HID; ++k) acc += comb[k] * W1[k * HID + t];
        o1[t] = acc > 0.0f ? acc : 0.0f;
    }
    __syncthreads();
    if (t < HID / 2) {
        float acc = b2[t];
        for (int k = 0; k < HID; ++k) acc += o1[k] * W2[k * (HID / 2) + t];
        o2[t] = acc > 0.0f ? acc : 0.0f;
    }
    __syncthreads();
    if (t == 0) {
        float acc = b3[0];
        for (int k = 0; k < HID / 2; ++k) acc += o2[k] * W3[k];
        out[0] = acc;
    }
}

// ---------------------------------------------------------------------------
extern "C" void kernel_launch(void* const* d_in, const int* in_sizes, int n_in,
                              void* d_out, int out_size, void* d_ws, size_t ws_size,
                              hipStream_t stream) {
    const float* x   = (const float*)d_in[0];
    const int*   ei  = (const int*)d_in[1];
    // d_in[2] = edge_attr : unused by the network (edge encoder output is dead code)
    const float* nW1 = (const float*)d_in[3];
    const float* nb1 = (const float*)d_in[4];
    const float* nW2 = (const float*)d_in[5];
    const float* nb2 = (const float*)d_in[6];
    // d_in[7..10] = edge MLP params : unused
    const float* cW[3] = { (const float*)d_in[11], (const float*)d_in[12], (const float*)d_in[13] };
    const float* cb[3] = { (const float*)d_in[14], (const float*)d_in[15], (const float*)d_in[16] };
    const float* oW1 = (const float*)d_in[17];
    const float* ob1 = (const float*)d_in[18];
    const float* oW2 = (const float*)d_in[19];
    const float* ob2 = (const float*)d_in[20];
    const float* oW3 = (const float*)d_in[21];
    const float* ob3 = (const float*)d_in[22];

    const int n  = in_sizes[0] / 4;
    const int nE = in_sizes[2];

    float* h    = (float*)d_ws;
    float* hw   = h   + (size_t)n * HID;
    float* agg  = hw  + (size_t)n * HID;
    float* deg  = agg + (size_t)n * HID;
    float* dinv = deg + n;
    int*   idx  = (int*)(dinv + n);

    const int B = 256;
    long long elemsN = (long long)n * HID;
    long long elemsE = (long long)nE * HID;

    // normalization terms + flag indices
    deg_init_kernel<<<(n + B - 1) / B, B, 0, stream>>>(deg, n);
    deg_accum_kernel<<<(nE + B - 1) / B, B, 0, stream>>>(ei, deg, nE);
    dinv_kernel<<<(n + B - 1) / B, B, 0, stream>>>(deg, dinv, n);
    idx_init_kernel<<<1, 32, 0, stream>>>(idx);
    find_idx_kernel<<<(n + B - 1) / B, B, 0, stream>>>(x, idx, n);

    // node encoder (WMMA)
    node_mlp_kernel<<<(n + 63) / 64, 128, 0, stream>>>(x, nW1, nb1, nW2, nb2, h, n);

    // 3 GCN layers
    for (int l = 0; l < 3; ++l) {
        gcn_linear_kernel<<<(n + 63) / 64, 128, 0, stream>>>(h, cW[l], hw, n);
        selfloop_kernel<<<(int)((elemsN + B - 1) / B), B, 0, stream>>>(hw, dinv, agg, n);
        scatter_kernel<<<(int)((elemsE + B - 1) / B), B, 0, stream>>>(hw, ei, dinv, agg, nE);
        update_kernel<<<(int)((elemsN + B - 1) / B), B, 0, stream>>>(h, agg, cb[l], n);
    }

    // readout
    readout_kernel<<<1, 128, 0, stream>>>(h, idx, oW1, ob1, oW2, ob2, oW3, ob3, (float*)d_out);
}